// Graph_Learn_Spatial_48945447305785
// MI455X (gfx1250) — compile-verified
//
#include <hip/hip_runtime.h>

// CDNA5 / gfx1250. wave32.
typedef __attribute__((ext_vector_type(2))) float v2f;
typedef __attribute__((ext_vector_type(8))) float v8f;

#define N_ 16
#define T_ 128
#define V_ 64
#define F_ 16

// -----------------------------------------------------------------------------
// Kernel 1: score[n,i,j] = (1/T) * sum_{t,f} a[f] * |x[n,t,i,f] - x[n,t,j,f]|
//
// GEMV via V_WMMA_F32_16X16X4_F32:
//   - M = 16 rows: i = ig*16 + m for a fixed j
//   - K = (t, f) flattened, consumed 4 at a time (one f-chunk of 4 per t)
//   - A[m,k] = (a[f]/T) * |x_i - x_j|   (weights folded into A)
//   - B = all-ones  -> every column of C equals the score (layout-agnostic)
//
// A-fragment layout (32-bit A 16x4, ISA 7.12.2):
//   lane L: M = L&15 ; VGPR0 -> K = (L>=16 ? 2 : 0), VGPR1 -> K+1
// C/D layout: lane L, VGPR r holds (M = r + 8*(L>=16), N = L&15).
// -----------------------------------------------------------------------------
__global__ __launch_bounds__(128)
void gls_score_kernel(const float* __restrict__ x,
                      const float* __restrict__ a_spatial,
                      float* __restrict__ score) {
  const int lane = threadIdx.x & 31;
  const int wave = blockIdx.x * 4 + (threadIdx.x >> 5);  // 0..4095
  const int j  = wave & (V_ - 1);
  const int ig = (wave >> 6) & 3;        // i-group (4 groups of 16 rows)
  const int n  = wave >> 8;

  const int m    = lane & 15;            // WMMA row
  const int ksub = (lane >> 4) << 1;     // 0 (lanes 0-15) or 2 (lanes 16-31)
  const int i    = ig * 16 + m;

  // Per-lane weights for each f-chunk: a[fc*4 + ksub + {0,1}] / T
  v2f wv[4];
#pragma unroll
  for (int fc = 0; fc < 4; ++fc) {
    wv[fc].x = a_spatial[fc * 4 + ksub + 0] * (1.0f / (float)T_);
    wv[fc].y = a_spatial[fc * 4 + ksub + 1] * (1.0f / (float)T_);
  }

  const v2f ones = {1.0f, 1.0f};
  v8f c = {};

  const float* xi_base = x + ((size_t)n * T_ * V_ + i) * F_;
  const float* xj_base = x + ((size_t)n * T_ * V_ + j) * F_;

  for (int t = 0; t < T_; ++t) {
    const float* xi = xi_base + (size_t)t * (V_ * F_);
    const float* xj = xj_base + (size_t)t * (V_ * F_);
#pragma unroll
    for (int fc = 0; fc < 4; ++fc) {
      const int f0 = fc * 4 + ksub;
      v2f pi = *(const v2f*)(xi + f0);   // global_load_b64
      v2f pj = *(const v2f*)(xj + f0);   // global_load_b64
      v2f av;
      av.x = wv[fc].x * fabsf(pi.x - pj.x);
      av.y = wv[fc].y * fabsf(pi.y - pj.y);
      // 8 args: (neg_a, A, neg_b, B, c_mod, C, reuse_a, reuse_b)
      c = __builtin_amdgcn_wmma_f32_16x16x4_f32(false, av, false, ones,
                                                (short)0, c, false, false);
    }
  }

  // Every column of C is identical; lane 0 holds M=0..7, lane 16 holds M=8..15.
  if ((lane & 15) == 0) {
    const int mBase = (lane >> 4) * 8;
    float* out = score + ((size_t)n * V_ + (size_t)ig * 16 + mBase) * V_ + j;
#pragma unroll
    for (int r = 0; r < 8; ++r) out[(size_t)r * V_] = c[r];
  }
}

// -----------------------------------------------------------------------------
// Kernel 2: out[n,i,j] = exp(relu(score)) / sum_i exp(relu(score[n,i,j]))
// One wave per (n, j); each lane handles 2 i-values; wave32 shuffle reduction.
// -----------------------------------------------------------------------------
__global__ __launch_bounds__(32)
void gls_norm_kernel(const float* __restrict__ score,
                     float* __restrict__ out) {
  const int lane = threadIdx.x;          // 0..31
  const int nj = blockIdx.x;             // 0..N*V-1
  const int j = nj & (V_ - 1);
  const int n = nj >> 6;

  const float* s = score + (size_t)n * V_ * V_ + j;
  const int i0 = lane * 2;

  float e0 = expf(fmaxf(s[(size_t)(i0 + 0) * V_], 0.0f));
  float e1 = expf(fmaxf(s[(size_t)(i0 + 1) * V_], 0.0f));
  float p = e0 + e1;
#pragma unroll
  for (int off = 16; off > 0; off >>= 1)
    p += __shfl_xor(p, off, 32);
  const float inv = 1.0f / p;

  float* o = out + (size_t)n * V_ * V_ + j;
  o[(size_t)(i0 + 0) * V_] = e0 * inv;
  o[(size_t)(i0 + 1) * V_] = e1 * inv;
}

// -----------------------------------------------------------------------------
extern "C" void kernel_launch(void* const* d_in, const int* in_sizes, int n_in,
                              void* d_out, int out_size, void* d_ws, size_t ws_size,
                              hipStream_t stream) {
  const float* x = (const float*)d_in[0];        // [N,T,V,F] f32
  const float* a = (const float*)d_in[1];        // [F,1] f32
  float* out   = (float*)d_out;                  // [N,V,V] f32
  float* score = (float*)d_ws;                   // [N,V,V] f32 scratch (256 KB)

  // 4096 waves total, 4 waves (128 threads) per block -> 1024 blocks
  gls_score_kernel<<<1024, 128, 0, stream>>>(x, a, score);
  gls_norm_kernel<<<N_ * V_, 32, 0, stream>>>(score, out);
}